// Network_p2_c3_321_21234318312194
// MI455X (gfx1250) — compile-verified
//
#include <hip/hip_runtime.h>

namespace {

constexpr int   kResUp        = 224;
constexpr float kScaleUp      = 223.0f;   // RES_UP - 1
constexpr int   kResDn        = 8;
constexpr float kScaleDn      = 7.0f;     // RES_DN - 1
constexpr int   kLutTexels    = kResDn * kResDn * kResDn;      // 512
constexpr int   kLutBytes     = kLutTexels * 16;               // 8 KB
constexpr int   kLutFloats    = kLutBytes / 4;                 // 2048
constexpr int   k2dFloats     = kResUp * kResUp * 3;           // 150528
constexpr int   kThreads      = 256;
constexpr int   kPtsPerThread = 4;

// Staging mode for the 8 KB LDS LUT.
constexpr int kStageTDM   = 0;   // Tensor Data Mover (descriptor DMA)
constexpr int kStageAsync = 1;   // global_load_async_to_lds_b128 loop
constexpr int kStageValu  = 2;   // plain loads + ds_store (no scratch needed)

typedef unsigned int u32x4 __attribute__((ext_vector_type(4)));
typedef unsigned int u32x8 __attribute__((ext_vector_type(8)));

struct F3 { float x, y, z; };
struct Row6 { float f[6]; };   // one bilinear row: c0.xyz ‖ c1.xyz (24 B contiguous)

__device__ __forceinline__ float fract1(float t) { return t - floorf(t); }

__device__ __forceinline__ F3 lerp3(F3 a, F3 b, float t) {
  return { fmaf(t, b.x - a.x, a.x),
           fmaf(t, b.y - a.y, a.y),
           fmaf(t, b.z - a.z, a.z) };
}

template <bool kFract>
__device__ __forceinline__ F3 f3_from(const float* f) {
  if constexpr (kFract)
    return { fract1(f[0]), fract1(f[1]), fract1(f[2]) };
  else
    return { f[0], f[1], f[2] };
}

__device__ __forceinline__ F3 lut_ld(const float4* lut, int i) {
  float4 c = lut[i];               // single ds_load_b128 per texel
  return { c.x, c.y, c.z };
}

// kFract2D: apply nonlin() to 2D-table values at gather time (true when the
// table was NOT pre-fracted into scratch).
template <bool kFract2D>
__device__ __forceinline__ F3 eval_point(float xx, float yy,
                                         const float* __restrict__ t2d,
                                         const float4* lut) {
  // ---- bilinear lookup into (fract of) table2d, 224x224x3, global/L2 ----
  float u = xx * kScaleUp;
  float v = yy * kScaleUp;
  int u0 = (int)floorf(u);
  int v0 = (int)floorf(v);
  u0 = min(max(u0, 0), kResUp - 2);
  v0 = min(max(v0, 0), kResUp - 2);
  float fu = u - (float)u0;
  float fv = v - (float)v0;
  int r0 = (u0 * kResUp + v0) * 3;
  int r1 = r0 + kResUp * 3;
  // Two wide 24B row gathers (unaligned b128/b96 vector loads).
  Row6 rowA = *reinterpret_cast<const Row6*>(t2d + r0);
  Row6 rowB = *reinterpret_cast<const Row6*>(t2d + r1);
  F3 c00 = f3_from<kFract2D>(rowA.f);
  F3 c01 = f3_from<kFract2D>(rowA.f + 3);
  F3 c10 = f3_from<kFract2D>(rowB.f);
  F3 c11 = f3_from<kFract2D>(rowB.f + 3);
  F3 key = lerp3(lerp3(c00, c01, fv), lerp3(c10, c11, fv), fu);

  // ---- trilinear lookup into fract(table3d), 8x8x8x3, LDS (float4 texels) ----
  float ku = key.x * kScaleDn;
  float kv = key.y * kScaleDn;
  float kw = key.z * kScaleDn;
  int a0 = min(max((int)floorf(ku), 0), kResDn - 2);
  int b0 = min(max((int)floorf(kv), 0), kResDn - 2);
  int d0 = min(max((int)floorf(kw), 0), kResDn - 2);
  float gu = ku - (float)a0;
  float gv = kv - (float)b0;
  float gw = kw - (float)d0;
  int base = (a0 * kResDn + b0) * kResDn + d0;
  F3 t000 = lut_ld(lut, base);
  F3 t001 = lut_ld(lut, base + 1);
  F3 t010 = lut_ld(lut, base + kResDn);
  F3 t011 = lut_ld(lut, base + kResDn + 1);
  F3 t100 = lut_ld(lut, base + kResDn * kResDn);
  F3 t101 = lut_ld(lut, base + kResDn * kResDn + 1);
  F3 t110 = lut_ld(lut, base + kResDn * kResDn + kResDn);
  F3 t111 = lut_ld(lut, base + kResDn * kResDn + kResDn + 1);
  F3 e0 = lerp3(lerp3(t000, t001, gw), lerp3(t010, t011, gw), gv);
  F3 e1 = lerp3(lerp3(t100, t101, gw), lerp3(t110, t111, gw), gv);
  return lerp3(e0, e1, gu);
}

} // namespace

// Pass 1a: fract(table3d), pad each 3-float texel to a 16B slot at ws[0..8KB).
__global__ void __launch_bounds__(kThreads)
prep3d_kernel(const float* __restrict__ t3d, float4* __restrict__ ws4) {
  int t = blockIdx.x * blockDim.x + threadIdx.x;
  if (t < kLutTexels) {
    float a = t3d[t * 3 + 0];
    float b = t3d[t * 3 + 1];
    float c = t3d[t * 3 + 2];
    ws4[t] = make_float4(fract1(a), fract1(b), fract1(c), 0.0f);
  }
}

// Pass 1b: fract(table2d) streamed float4-wise into ws[8KB ...).
__global__ void __launch_bounds__(kThreads)
prep2d_kernel(const float4* __restrict__ t2d4, float4* __restrict__ ws2d4) {
  int t = blockIdx.x * blockDim.x + threadIdx.x;
  if (t < k2dFloats / 4) {
    float4 v = t2d4[t];
    ws2d4[t] = make_float4(fract1(v.x), fract1(v.y), fract1(v.z), fract1(v.w));
  }
}

// Pass 2: fused bilinear->trilinear evaluation.
template <int kStage, bool kFract2D>
__global__ void __launch_bounds__(kThreads)
fused_grid_kernel(const float* __restrict__ x,
                  const float* __restrict__ t2d,   // raw or pre-fracted table2d
                  const float* __restrict__ t3d,   // raw table3d (fallback path)
                  const float4* __restrict__ ws4,  // pre-fracted padded 3D LUT
                  float* __restrict__ out,
                  long long npts) {
  __shared__ float4 lut[kLutTexels];   // 8 KB

  if constexpr (kStage == kStageTDM) {
    // Stage the 8 KB LUT with one Tensor Data Mover descriptor (wave 0 only;
    // wave-uniform branch so EXEC-ignoring TDM can't issue from other waves).
    if (threadIdx.x < 32) {
      unsigned ldsAddr = (unsigned)(size_t)(void*)&lut[0];
      unsigned long long ga = (unsigned long long)(size_t)ws4;
      // D# group 0: count=1, lds_addr, global_addr[56:0], type=2.
      u32x4 g0 = { 0x1u,
                   ldsAddr,
                   (unsigned)(ga & 0xFFFFFFFFull),
                   (unsigned)((ga >> 32) & 0x01FFFFFFull) | (2u << 30) };
      // D# group 1: data_size=4B, tensor_dim0=2048, tensor_dim1=1,
      // tile_dim0=2048, tile_dim1=1, dim0_stride=2048, dim1_stride=2048.
      u32x8 g1 = { 0x00020000u,            // workgroup_mask=0, data_size=2 (4B)
                   0x08000000u,            // tensor_dim0 lo16 << 16 (2048)
                   0x00010000u,            // tensor_dim0 hi16=0, tensor_dim1 lo16=1
                   0x08000000u,            // tensor_dim1 hi16=0, tile_dim0=2048
                   0x00000001u,            // tile_dim1=1, tile_dim2=0
                   0x00000800u,            // dim0_stride lo32 = 2048
                   0x08000000u,            // dim0_stride hi16=0, dim1_stride lo16=2048
                   0x00000000u };          // dim1_stride hi32 = 0
      asm volatile("tensor_load_to_lds %0, %1" :: "s"(g0), "s"(g1) : "memory");
      __builtin_amdgcn_s_wait_tensorcnt(0);
    }
  } else if constexpr (kStage == kStageAsync) {
    // Stage prefracted 3D LUT into LDS with async global->LDS b128 copies.
    unsigned ldsBase = (unsigned)(size_t)(void*)&lut[0];
    for (int c = threadIdx.x; c < kLutTexels; c += kThreads) {
      unsigned long long src = (unsigned long long)(size_t)(ws4 + c);
      unsigned dst = ldsBase + (unsigned)c * 16u;
      asm volatile("global_load_async_to_lds_b128 %0, %1, off"
                   :: "v"(dst), "v"(src) : "memory");
    }
    asm volatile("s_wait_asynccnt 0x0" ::: "memory");
  } else {
    // Fallback: VALU staging straight from table3d (fract + pad on the fly).
    for (int c = threadIdx.x; c < kLutTexels; c += kThreads) {
      float a = t3d[c * 3 + 0];
      float b = t3d[c * 3 + 1];
      float d = t3d[c * 3 + 2];
      lut[c] = make_float4(fract1(a), fract1(b), fract1(d), 0.0f);
    }
  }
  __syncthreads();

  long long gid = (long long)blockIdx.x * kThreads + threadIdx.x;
  long long p0  = gid * kPtsPerThread;
  if (p0 + kPtsPerThread <= npts) {
    // Vector path: 4 points/thread -> b128 streaming loads/stores.
    const float4* x4 = reinterpret_cast<const float4*>(x);
    float4 xa = x4[gid * 2 + 0];
    float4 xb = x4[gid * 2 + 1];
    float px[4] = { xa.x, xa.z, xb.x, xb.z };
    float py[4] = { xa.y, xa.w, xb.y, xb.w };
    float4 r[3];
    float* rf = reinterpret_cast<float*>(r);
#pragma unroll
    for (int p = 0; p < kPtsPerThread; ++p) {
      F3 o = eval_point<kFract2D>(px[p], py[p], t2d, lut);
      rf[p * 3 + 0] = o.x;
      rf[p * 3 + 1] = o.y;
      rf[p * 3 + 2] = o.z;
    }
    float4* o4 = reinterpret_cast<float4*>(out) + gid * 3;
    o4[0] = r[0];
    o4[1] = r[1];
    o4[2] = r[2];
  } else {
    // Scalar tail (not taken for N = 8388608, kept for generality).
    for (long long p = p0; p < npts; ++p) {
      F3 o = eval_point<kFract2D>(x[2 * p], x[2 * p + 1], t2d, lut);
      out[3 * p + 0] = o.x;
      out[3 * p + 1] = o.y;
      out[3 * p + 2] = o.z;
    }
  }
}

extern "C" void kernel_launch(void* const* d_in, const int* in_sizes, int n_in,
                              void* d_out, int out_size, void* d_ws, size_t ws_size,
                              hipStream_t stream) {
  const float* x   = (const float*)d_in[0];   // (N,2)
  const float* t2d = (const float*)d_in[1];   // (224,224,3)
  const float* t3d = (const float*)d_in[2];   // (8,8,8,3)
  float* out = (float*)d_out;                 // (N,3)

  long long npts = (long long)in_sizes[0] / 2;
  long long nthr = (npts + kPtsPerThread - 1) / kPtsPerThread;
  int blocks = (int)((nthr + kThreads - 1) / kThreads);

  const size_t wsNeed3d   = (size_t)kLutBytes;                          // 8 KB
  const size_t wsNeedFull = (size_t)kLutBytes + (size_t)k2dFloats * 4;  // ~610 KB

  if (ws_size >= wsNeedFull) {
    // Tier 1: pre-fract both tables into scratch; TDM-staged LDS LUT.
    float4* ws4   = (float4*)d_ws;                       // padded 3D LUT
    float4* ws2d4 = (float4*)((char*)d_ws + kLutBytes);  // fract(table2d)
    int pb3 = (kLutTexels + kThreads - 1) / kThreads;
    int pb2 = (k2dFloats / 4 + kThreads - 1) / kThreads;
    prep3d_kernel<<<pb3, kThreads, 0, stream>>>(t3d, ws4);
    prep2d_kernel<<<pb2, kThreads, 0, stream>>>(
        reinterpret_cast<const float4*>(t2d), ws2d4);
    fused_grid_kernel<kStageTDM, false><<<blocks, kThreads, 0, stream>>>(
        x, (const float*)ws2d4, t3d, ws4, out, npts);
  } else if (ws_size >= wsNeed3d) {
    // Tier 2: pre-fract only the 3D LUT; async-b128 staging; fract 2D at gather.
    float4* ws4 = (float4*)d_ws;
    int pb3 = (kLutTexels + kThreads - 1) / kThreads;
    prep3d_kernel<<<pb3, kThreads, 0, stream>>>(t3d, ws4);
    fused_grid_kernel<kStageAsync, true><<<blocks, kThreads, 0, stream>>>(
        x, t2d, t3d, ws4, out, npts);
  } else {
    // Tier 3: no scratch at all.
    fused_grid_kernel<kStageValu, true><<<blocks, kThreads, 0, stream>>>(
        x, t2d, t3d, nullptr, out, npts);
  }
}